// FFT_33827162423767
// MI455X (gfx1250) — compile-verified
//
#include <hip/hip_runtime.h>

typedef __attribute__((ext_vector_type(2))) float v2f;
typedef __attribute__((ext_vector_type(4))) float v4f;
typedef __attribute__((ext_vector_type(8))) float v8f;

// T[m][k]: coefficient of flat input element k (x*4+y*2+r, r=0 real / r=1 imag)
// in flat output element m (u*4+v; v<2 -> Re(xf[u][v]), v>=2 -> Im(xf[u][v-2]))
// of the 4x2 complex FFT2. All entries are in {0, +1, -1}.
__device__ __forceinline__ float fft_t(int m, int k) {
    int u    = m >> 2;
    int v    = m & 3;
    int comp = v >> 1;   // 0 = real output, 1 = imag output
    int vp   = v & 1;    // frequency along length-2 axis
    int x    = k >> 2;
    int y    = (k >> 1) & 1;
    int r    = k & 1;    // 0 = real input, 1 = imag input
    int phi  = (u * x) & 3;                       // e^{-i*pi*phi/2} in {1,-i,-1,i}
    float re4 = (phi == 0) ? 1.0f : (phi == 2) ? -1.0f : 0.0f;
    float im4 = (phi == 1) ? -1.0f : (phi == 3) ? 1.0f : 0.0f;
    float s   = (vp & y) ? -1.0f : 1.0f;          // (-1)^{v'*y}
    if (comp == 0) return (r == 0) ? s * re4 : -s * im4;  // Re(W*X)
    else           return (r == 0) ? s * im4 :  s * re4;  // Im(W*X)
}

// Each wave transforms tiles of 16 groups (16 consecutive floats each):
//   D(16x16) = T(16x16) * X(16x16)  via 4x V_WMMA_F32_16X16X4_F32.
// Loop counter is forced wave-uniform (readfirstlane) so control flow is
// scalar and EXEC provably stays all-1s through the WMMAs (ISA requirement).
__global__ __launch_bounds__(256) void fft2_wmma_kernel(
    const float* __restrict__ in, float* __restrict__ out, int numTiles) {
    const int lane          = threadIdx.x & 31;
    const int wavesPerBlock = blockDim.x >> 5;

    // wave-uniform scalars
    const int wave = __builtin_amdgcn_readfirstlane(
        blockIdx.x * wavesPerBlock + (threadIdx.x >> 5));
    const int numWaves = gridDim.x * wavesPerBlock;

    const int m = lane & 15;   // A-matrix row / B,D column (group within tile)
    const int h = lane >> 4;   // lane half selects K (A,B) / M (D) upper half

    // A-matrix (constant transform), chunk c covers K = 4c..4c+3.
    // ISA layout: lane L, VGPR v -> T[L&15][4c + 2*(L>>4) + v]
    v2f a[4];
#pragma unroll
    for (int c = 0; c < 4; ++c) {
        a[c][0] = fft_t(m, 4 * c + 2 * h + 0);
        a[c][1] = fft_t(m, 4 * c + 2 * h + 1);
    }

    // per-lane float offsets within a 1KiB tile (loop-invariant)
    const int srcLaneOff = m * 16 + 2 * h;   // floats
    const int dstLaneOff = m * 16 + 8 * h;   // floats

    for (int tile = wave; tile < numTiles; tile += numWaves) {
        const float* tbase = in + (size_t)tile * 256;

        if (tile + numWaves < numTiles) {
            // scalar-guarded stream-ahead: wave covers next 2KiB at 64B steps
            __builtin_prefetch(
                (const char*)(in + (size_t)(tile + numWaves) * 256) + lane * 64,
                0, 3);
        }

        // B-matrix: lane L, VGPR v -> X[4c + 2*(L>>4) + v][L&15]
        // = input floats (4c+2h, 4c+2h+1) of group m -> one b64 load per chunk
        const float* src = tbase + srcLaneOff;
        v2f b[4];
#pragma unroll
        for (int c = 0; c < 4; ++c)
            b[c] = __builtin_nontemporal_load((const v2f*)(src + 4 * c));

        v8f d = {};
#pragma unroll
        for (int c = 0; c < 4; ++c)
            d = __builtin_amdgcn_wmma_f32_16x16x4_f32(
                    /*neg_a=*/false, a[c], /*neg_b=*/false, b[c],
                    /*c_mod=*/(short)0, d, /*reuse_a=*/false, /*reuse_b=*/false);

        // D: lane L, VGPR v -> out element (v + 8h) of group m
        // -> 32 contiguous bytes per lane, wave writes 1KiB contiguous
        float* dst = out + (size_t)tile * 256 + dstLaneOff;
        v4f lo = {d[0], d[1], d[2], d[3]};
        v4f hi = {d[4], d[5], d[6], d[7]};
        __builtin_nontemporal_store(lo, (v4f*)dst);
        __builtin_nontemporal_store(hi, (v4f*)(dst + 4));
    }
}

extern "C" void kernel_launch(void* const* d_in, const int* in_sizes, int n_in,
                              void* d_out, int out_size, void* d_ws, size_t ws_size,
                              hipStream_t stream) {
    const float* x = (const float*)d_in[0];
    float* out = (float*)d_out;

    const int numTiles = in_sizes[0] / 256;  // 16 groups of 16 floats per tile

    const int threads = 256;                 // 8 waves per block
    const int wavesPerBlock = threads / 32;
    int blocksNeeded = (numTiles + wavesPerBlock - 1) / wavesPerBlock;
    int blocks = blocksNeeded < 8192 ? blocksNeeded : 8192;  // grid-stride

    fft2_wmma_kernel<<<blocks, threads, 0, stream>>>(x, out, numTiles);
}